// TimestepNorm_43585328119922
// MI455X (gfx1250) — compile-verified
//
#include <hip/hip_runtime.h>
#include <stdint.h>

// ---------------------------------------------------------------------------
// TimestepNorm for MI455X (gfx1250): decoupled Welford scan.
// B=16, L=4096, D=1024, fp32. Memory-bound (~790 MB traffic => ~34 us @ 23.3 TB/s).
// Uses CDNA5 Tensor Data Mover (tensor_load_to_lds) to stage x tiles in LDS.
// ---------------------------------------------------------------------------

#define TN_B  16
#define TN_L  4096
#define TN_D  1024
#define TN_CT 64                 // chunk length (timesteps)
#define TN_NC (TN_L / TN_CT)     // 64 chunks
#define TN_TD 256                // channel-tile width == blockDim.x
#define TN_NDT (TN_D / TN_TD)    // 4 d-tiles
#define TN_EPS 1e-5f

typedef __attribute__((ext_vector_type(4))) unsigned int u32x4_t;
typedef __attribute__((ext_vector_type(8))) int          i32x8_t;
typedef __attribute__((ext_vector_type(4))) int          i32x4_t;

// Issue one TDM 2D tile load: global (tile_t rows x tile_d elems, row stride
// stride_elems, 4-byte elements) -> LDS at byte offset lds_off (packed).
// Descriptor layout per CDNA5 ISA ch.8 (D# group0 / group1).
__device__ __forceinline__ void tdm_load_tile_f32(const float* gptr,
                                                  unsigned lds_off,
                                                  unsigned tile_d,
                                                  unsigned tile_t,
                                                  unsigned stride_elems)
{
#if defined(__HIP_DEVICE_COMPILE__)
    unsigned long long ga = (unsigned long long)(uintptr_t)gptr;

    u32x4_t g0;
    g0[0] = 1u;                                   // count=1 (valid descriptor)
    g0[1] = lds_off;                              // lds_addr [63:32]
    g0[2] = (unsigned)(ga & 0xFFFFFFFFu);         // global_addr low
    g0[3] = ((unsigned)(ga >> 32) & 0x01FFFFFFu)  // global_addr [56:32]
          | 0x80000000u;                          // type=2 ("image") [127:126]

    i32x8_t g1;
    g1[0] = 0x00020000;                                        // data_size=2 (4B), no multicast/pad/iterate
    g1[1] = (int)((tile_d & 0xFFFFu) << 16);                   // tensor_dim0 low16 @ [63:48]
    g1[2] = (int)((tile_d >> 16) | ((tile_t & 0xFFFFu) << 16));// tensor_dim0 hi / tensor_dim1 low
    g1[3] = (int)((tile_t >> 16) | ((tile_d & 0xFFFFu) << 16));// tensor_dim1 hi / tile_dim0
    g1[4] = (int)(tile_t & 0xFFFFu);                           // tile_dim1, tile_dim2=0
    g1[5] = (int)stride_elems;                                 // tensor_dim0_stride low32
    g1[6] = 0;                                                 // stride hi / dim1_stride low
    g1[7] = 0;

    i32x4_t z4 = {0, 0, 0, 0};
    i32x8_t z8 = {0, 0, 0, 0, 0, 0, 0, 0};
    // clang-23 / therock form: (g0, g1, g2, g3, g4, cpol)
    __builtin_amdgcn_tensor_load_to_lds(g0, g1, z4, z4, z8, 0);
    __builtin_amdgcn_s_wait_tensorcnt(0);
#else
    (void)gptr; (void)lds_off; (void)tile_d; (void)tile_t; (void)stride_elems;
#endif
}

// ---------------------------------------------------------------------------
// Pass 1: per-chunk masked statistics (n, mean, M2) for each (b, chunk, d).
// Grid: B*NC*NDT blocks of 256 threads.
// ---------------------------------------------------------------------------
__global__ void __launch_bounds__(TN_TD)
k_chunk_stats(const float* __restrict__ x,
              const unsigned char* __restrict__ pad,   // padding_mask, 1 byte / elem
              float* __restrict__ cmean,
              float* __restrict__ cM2,
              float* __restrict__ cn)
{
    __shared__ float tile[TN_CT * TN_TD];   // 64 KB
    __shared__ float smask[TN_CT];

    const int tid = threadIdx.x;
    const int bid = blockIdx.x;
    const int dt  = bid % TN_NDT;
    const int c   = (bid / TN_NDT) % TN_NC;
    const int b   = bid / (TN_NDT * TN_NC);
    const int t0  = c * TN_CT;
    const int d0  = dt * TN_TD;

    if (tid < TN_CT)
        smask[tid] = pad[(size_t)b * TN_L + t0 + tid] ? 0.0f : 1.0f;

    if (tid < 32) {   // wave 0 issues the DMA (uniform scalar operands)
        tdm_load_tile_f32(x + ((size_t)b * TN_L + t0) * TN_D + d0,
                          (unsigned)(uintptr_t)(void*)tile,
                          TN_TD, TN_CT, TN_D);
    }
    __syncthreads();

    float S1 = 0.0f, S2 = 0.0f, n = 0.0f;
#pragma unroll 8
    for (int t = 0; t < TN_CT; ++t) {
        const float m = smask[t];
        const float v = tile[t * TN_TD + tid];
        const float mv = m * v;
        n  += m;
        S1 += mv;
        S2  = fmaf(mv, v, S2);
    }

    float mean = 0.0f, M2 = 0.0f;
    if (n > 0.0f) {
        mean = S1 / n;
        M2   = fmaf(-S1, mean, S2);      // sum(x^2) - sum(x)^2/n
        if (M2 < 0.0f) M2 = 0.0f;
    }

    const size_t idx = ((size_t)(b * TN_NC + c)) * TN_D + d0 + tid;
    cmean[idx] = mean;
    cM2[idx]   = M2;
    if (d0 + tid == 0)
        cn[b * TN_NC + c] = n;
}

// ---------------------------------------------------------------------------
// Pass 2: sequential prefix-merge of chunk stats per (b, d); converts
// cmean/cM2 IN PLACE into exclusive prefixes, records exclusive counts,
// and emits final (count, mean, var) outputs.
// Grid: (B*D)/256 blocks of 256 threads.
// ---------------------------------------------------------------------------
__global__ void __launch_bounds__(256)
k_prefix(const int*   __restrict__ prev_count,
         const float* __restrict__ prev_mean,
         const float* __restrict__ prev_var,
         float* __restrict__ cmean,   // in: chunk stats, out: exclusive prefix
         float* __restrict__ cM2,
         const float* __restrict__ cn,
         float* __restrict__ en,      // exclusive count per (b, chunk)
         float* __restrict__ outCount,
         float* __restrict__ outMean,
         float* __restrict__ outVar)
{
    const int gid = blockIdx.x * blockDim.x + threadIdx.x;   // [0, B*D)
    const int b = gid / TN_D;
    const int d = gid % TN_D;

    float n    = (float)prev_count[b];      // >= 1
    float mean = prev_mean[gid];
    float M2   = prev_var[gid] * n;

    for (int c = 0; c < TN_NC; ++c) {
        const size_t idx = ((size_t)(b * TN_NC + c)) * TN_D + d;
        const float nc = cn[b * TN_NC + c];
        const float mc = cmean[idx];
        const float Mc = cM2[idx];

        if (d == 0) en[b * TN_NC + c] = n;
        cmean[idx] = mean;                  // exclusive prefix
        cM2[idx]   = M2;

        if (nc > 0.0f) {                    // Welford/Chan parallel merge
            const float nn    = n + nc;
            const float delta = mc - mean;
            const float r     = nc / nn;
            mean = fmaf(delta, r, mean);
            M2   = M2 + Mc + delta * delta * (n * r);
            n    = nn;
        }
    }

    outMean[gid] = mean;
    outVar[gid]  = M2 / n;
    if (d == 0) outCount[b] = n;
}

// ---------------------------------------------------------------------------
// Pass 3: replay exact per-step Welford recurrence inside each chunk starting
// from the exclusive prefix; produce y. Grid: B*NC*NDT blocks of 256.
// ---------------------------------------------------------------------------
__global__ void __launch_bounds__(TN_TD)
k_apply(const float* __restrict__ x,
        const unsigned char* __restrict__ pad,
        const float* __restrict__ weight,
        const float* __restrict__ bias,
        const float* __restrict__ emean,
        const float* __restrict__ eM2,
        const float* __restrict__ en,
        float* __restrict__ y)
{
    __shared__ float tile[TN_CT * TN_TD];   // 64 KB
    __shared__ float smask[TN_CT];

    const int tid = threadIdx.x;
    const int bid = blockIdx.x;
    const int dt  = bid % TN_NDT;
    const int c   = (bid / TN_NDT) % TN_NC;
    const int b   = bid / (TN_NDT * TN_NC);
    const int t0  = c * TN_CT;
    const int d0  = dt * TN_TD;
    const int d   = d0 + tid;

    if (tid < TN_CT)
        smask[tid] = pad[(size_t)b * TN_L + t0 + tid] ? 0.0f : 1.0f;

    if (tid < 32) {
        tdm_load_tile_f32(x + ((size_t)b * TN_L + t0) * TN_D + d0,
                          (unsigned)(uintptr_t)(void*)tile,
                          TN_TD, TN_CT, TN_D);
    }
    __syncthreads();

    const float gamma = weight[d] + 1.0f;
    const float beta  = bias[d];

    const size_t pidx = ((size_t)(b * TN_NC + c)) * TN_D + d;
    float n    = en[b * TN_NC + c];
    float mean = emean[pidx];
    float M2   = eM2[pidx];

    const size_t ybase = ((size_t)b * TN_L + t0) * TN_D + d;
#pragma unroll 4
    for (int t = 0; t < TN_CT; ++t) {
        const float m  = smask[t];          // uniform across the block
        const float xv = tile[t * TN_TD + tid];
        float out = 0.0f;
        if (m > 0.0f) {
            n += 1.0f;
            const float delta = xv - mean;
            mean += delta / n;
            const float d2 = xv - mean;
            M2 = fmaf(delta, d2, M2);
            const float var = M2 / n;
            out = fmaf(gamma * d2, rsqrtf(var + TN_EPS), beta);
        }
        y[ybase + (size_t)t * TN_D] = out;
    }
}

// ---------------------------------------------------------------------------
// Host launcher. Input order: x, prev_count, prev_mean, prev_var, weight,
// bias, padding_mask. Output: y | count | mean | var (flat concat).
// ---------------------------------------------------------------------------
extern "C" void kernel_launch(void* const* d_in, const int* in_sizes, int n_in,
                              void* d_out, int out_size, void* d_ws, size_t ws_size,
                              hipStream_t stream)
{
    const float*         x      = (const float*)d_in[0];
    const int*           pcount = (const int*)d_in[1];
    const float*         pmean  = (const float*)d_in[2];
    const float*         pvar   = (const float*)d_in[3];
    const float*         weight = (const float*)d_in[4];
    const float*         bias   = (const float*)d_in[5];
    const unsigned char* pad    = (const unsigned char*)d_in[6];

    float* out      = (float*)d_out;
    float* yOut     = out;                                          // B*L*D
    float* countOut = out + (size_t)TN_B * TN_L * TN_D;             // B
    float* meanOut  = countOut + TN_B;                              // B*D
    float* varOut   = meanOut + (size_t)TN_B * TN_D;                // B*D

    // Workspace: cmean | cM2 | cn | en
    const size_t statElems = (size_t)TN_B * TN_NC * TN_D;           // 1M floats
    float* cmean = (float*)d_ws;
    float* cM2   = cmean + statElems;
    float* cn    = cM2 + statElems;
    float* en    = cn + (size_t)TN_B * TN_NC;

    const int tileBlocks = TN_B * TN_NC * TN_NDT;   // 4096

    k_chunk_stats<<<tileBlocks, TN_TD, 0, stream>>>(x, pad, cmean, cM2, cn);

    k_prefix<<<(TN_B * TN_D) / 256, 256, 0, stream>>>(
        pcount, pmean, pvar, cmean, cM2, cn, en, countOut, meanOut, varOut);

    k_apply<<<tileBlocks, TN_TD, 0, stream>>>(
        x, pad, weight, bias, cmean, cM2, en, yOut);
}